// HiPPO_LegS_55138790146488
// MI455X (gfx1250) — compile-verified
//
#include <hip/hip_runtime.h>

typedef __attribute__((ext_vector_type(2))) float v2f;
typedef __attribute__((ext_vector_type(4))) float v4f;
typedef __attribute__((ext_vector_type(8))) float v8f;

#define N_DIM 256
#define B_DIM 64
#define L_DIM 1024
#define CHUNK_LEN 16
#define N_CHUNKS 64

__device__ __forceinline__ v8f wmma_f32_16x16x4(v2f a, v2f b, v8f c) {
  // D = A(16x4) * B(4x16) + C(16x16), all fp32, wave32
  return __builtin_amdgcn_wmma_f32_16x16x4_f32(
      /*neg_a=*/false, a, /*neg_b=*/false, b,
      /*c_mod=*/(short)0, c, /*reuse_a=*/false, /*reuse_b=*/false);
}

__device__ __forceinline__ void store8(float* dst, const v8f& a) {
  v4f lo = {a[0], a[1], a[2], a[3]};
  v4f hi = {a[4], a[5], a[6], a[7]};
  *reinterpret_cast<v4f*>(dst) = lo;
  *reinterpret_cast<v4f*>(dst + 4) = hi;
}

__device__ __forceinline__ v8f load8(const float* src) {
  v4f lo = *reinterpret_cast<const v4f*>(src);
  v4f hi = *reinterpret_cast<const v4f*>(src + 4);
  v8f r;
  r[0] = lo[0]; r[1] = lo[1]; r[2] = lo[2]; r[3] = lo[3];
  r[4] = hi[0]; r[5] = hi[1]; r[6] = hi[2]; r[7] = hi[3];
  return r;
}

// ---------------------------------------------------------------------------
// Pass 1: per chunk, evolve augmented state (256 identity cols -> P_k,
// 64 input-driven cols -> e_k) through CHUNK_LEN steps.
// Grid: 64 chunks x 2 column-groups; 1024 threads (32 waves).
// LDS state S[160][256] = 160 KB, layout [local_col][n].
// ---------------------------------------------------------------------------
__global__ __launch_bounds__(1024) void hippo_pass1(
    const float* __restrict__ A_st, const float* __restrict__ B_st,
    const float* __restrict__ inp, float* __restrict__ P_ws,
    float* __restrict__ e_ws) {
  __shared__ float S[160 * N_DIM];
  const int chunk = blockIdx.x >> 1;
  const int grp = blockIdx.x & 1;
  const int tid = threadIdx.x;
  const int w = tid >> 5;
  const int lane = tid & 31;
  const int half = lane >> 4;
  const int lm = lane & 15;
  const int mt = w & 15;           // output-row tile (M)
  const int jt0 = (w >> 4) * 5;    // 5 local column tiles per wave
  const int colbase = grp * 160;

  // Init: identity for global cols < 256, zero for the e-columns.
  for (int i = tid; i < 160 * N_DIM; i += 1024) {
    const int col = colbase + (i >> 8);
    const int n = i & 255;
    S[i] = (col == n) ? 1.0f : 0.0f;
  }
  __syncthreads();

  v8f acc[5];
  const int row = mt * 16 + lm;        // A-fragment row
  const int nb = mt * 16 + 8 * half;   // first M index held by this lane's acc

  for (int s = 0; s < CHUNK_LEN; ++s) {
    const int t = chunk * CHUNK_LEN + s;
    const float* __restrict__ A = A_st + (size_t)t * (N_DIM * N_DIM);

    const v4f bn0 = *reinterpret_cast<const v4f*>(B_st + t * N_DIM + nb);
    const v4f bn1 = *reinterpret_cast<const v4f*>(B_st + t * N_DIM + nb + 4);

#pragma unroll
    for (int j = 0; j < 5; ++j) {
      const int col = colbase + (jt0 + j) * 16 + lm;
      if (col >= N_DIM) {  // uniform per wave (tile-aligned)
        const float u = inp[t * B_DIM + (col - N_DIM)];
        acc[j][0] = bn0[0] * u; acc[j][1] = bn0[1] * u;
        acc[j][2] = bn0[2] * u; acc[j][3] = bn0[3] * u;
        acc[j][4] = bn1[0] * u; acc[j][5] = bn1[1] * u;
        acc[j][6] = bn1[2] * u; acc[j][7] = bn1[3] * u;
      } else {
        v8f z = {};
        acc[j] = z;
      }
    }

    if (t + 1 < L_DIM) {  // prefetch next step's A rows for this wave
      __builtin_prefetch(A + N_DIM * N_DIM + row * N_DIM, 0, 1);
      __builtin_prefetch(A + N_DIM * N_DIM + row * N_DIM + 128, 0, 1);
    }

#pragma unroll 4
    for (int kb = 0; kb < 64; ++kb) {
      const int k0 = kb * 4 + half * 2;
      const v2f a = *reinterpret_cast<const v2f*>(A + row * N_DIM + k0);
#pragma unroll
      for (int j = 0; j < 5; ++j) {
        const int lcol = (jt0 + j) * 16 + lm;
        const v2f b = *reinterpret_cast<const v2f*>(&S[lcol * N_DIM + k0]);
        acc[j] = wmma_f32_16x16x4(a, b, acc[j]);
      }
    }
    __syncthreads();  // all reads of S_old done
#pragma unroll
    for (int j = 0; j < 5; ++j) {
      const int lcol = (jt0 + j) * 16 + lm;
      store8(&S[lcol * N_DIM + nb], acc[j]);
    }
    __syncthreads();  // S_new visible
  }

  // Write P_k row-major [n][j] (so pass2 A-fragments are contiguous) and
  // e_k as [b][n].
#pragma unroll
  for (int j = 0; j < 5; ++j) {
    const int col = colbase + (jt0 + j) * 16 + lm;
    if (col < N_DIM) {
      float* dst = P_ws + ((size_t)chunk << 16);
#pragma unroll
      for (int v = 0; v < 8; ++v) dst[(nb + v) * N_DIM + col] = acc[j][v];
    } else {
      store8(e_ws + chunk * (B_DIM * N_DIM) + (col - N_DIM) * N_DIM + nb,
             acc[j]);
    }
  }
}

// ---------------------------------------------------------------------------
// Pass 2: serial scan over chunk summaries: s_{k+1} = P_k s_k + e_k.
// Records s_k (state at chunk start) for pass 3. One WG, 1024 threads.
// ---------------------------------------------------------------------------
__global__ __launch_bounds__(1024) void hippo_pass2(
    const float* __restrict__ P_ws, const float* __restrict__ e_ws,
    float* __restrict__ s_ws) {
  __shared__ float S[B_DIM * N_DIM];  // [b][n], 64 KB
  const int tid = threadIdx.x;
  const int w = tid >> 5;
  const int lane = tid & 31;
  const int half = lane >> 4;
  const int lm = lane & 15;
  const int mt = w >> 1;
  const int nt0 = (w & 1) * 2;  // 2 batch tiles per wave
  const int row = mt * 16 + lm;
  const int nb = mt * 16 + 8 * half;

  for (int i = tid; i < B_DIM * N_DIM; i += 1024) S[i] = 0.0f;
  __syncthreads();

  for (int k = 0; k < N_CHUNKS; ++k) {
    // Snapshot state at chunk start.
    for (int i = tid; i < B_DIM * N_DIM; i += 1024)
      s_ws[k * (B_DIM * N_DIM) + i] = S[i];

    const float* __restrict__ P = P_ws + ((size_t)k << 16);
    const float* __restrict__ e = e_ws + k * (B_DIM * N_DIM);

    v8f acc[2];
#pragma unroll
    for (int j = 0; j < 2; ++j) {
      const int b = (nt0 + j) * 16 + lm;
      acc[j] = load8(e + b * N_DIM + nb);
    }

#pragma unroll 4
    for (int kb = 0; kb < 64; ++kb) {
      const int k0 = kb * 4 + half * 2;
      const v2f a = *reinterpret_cast<const v2f*>(P + row * N_DIM + k0);
#pragma unroll
      for (int j = 0; j < 2; ++j) {
        const int b = (nt0 + j) * 16 + lm;
        const v2f bb = *reinterpret_cast<const v2f*>(&S[b * N_DIM + k0]);
        acc[j] = wmma_f32_16x16x4(a, bb, acc[j]);
      }
    }
    __syncthreads();
#pragma unroll
    for (int j = 0; j < 2; ++j) {
      const int b = (nt0 + j) * 16 + lm;
      store8(&S[b * N_DIM + nb], acc[j]);
    }
    __syncthreads();
  }
}

// ---------------------------------------------------------------------------
// Pass 3: replay each chunk from its correct initial state, writing outputs.
// Grid: 64 chunks; 1024 threads; LDS state [b][n] = 64 KB.
// ---------------------------------------------------------------------------
__global__ __launch_bounds__(1024) void hippo_pass3(
    const float* __restrict__ A_st, const float* __restrict__ B_st,
    const float* __restrict__ inp, const float* __restrict__ s_ws,
    float* __restrict__ out) {
  __shared__ float S[B_DIM * N_DIM];
  const int chunk = blockIdx.x;
  const int tid = threadIdx.x;
  const int w = tid >> 5;
  const int lane = tid & 31;
  const int half = lane >> 4;
  const int lm = lane & 15;
  const int mt = w >> 1;
  const int nt0 = (w & 1) * 2;
  const int row = mt * 16 + lm;
  const int nb = mt * 16 + 8 * half;

  for (int i = tid; i < B_DIM * N_DIM; i += 1024)
    S[i] = s_ws[chunk * (B_DIM * N_DIM) + i];
  __syncthreads();

  for (int s = 0; s < CHUNK_LEN; ++s) {
    const int t = chunk * CHUNK_LEN + s;
    const float* __restrict__ A = A_st + (size_t)t * (N_DIM * N_DIM);

    const v4f bn0 = *reinterpret_cast<const v4f*>(B_st + t * N_DIM + nb);
    const v4f bn1 = *reinterpret_cast<const v4f*>(B_st + t * N_DIM + nb + 4);

    v8f acc[2];
#pragma unroll
    for (int j = 0; j < 2; ++j) {
      const int b = (nt0 + j) * 16 + lm;
      const float u = inp[t * B_DIM + b];
      acc[j][0] = bn0[0] * u; acc[j][1] = bn0[1] * u;
      acc[j][2] = bn0[2] * u; acc[j][3] = bn0[3] * u;
      acc[j][4] = bn1[0] * u; acc[j][5] = bn1[1] * u;
      acc[j][6] = bn1[2] * u; acc[j][7] = bn1[3] * u;
    }

    if (t + 1 < L_DIM) {
      __builtin_prefetch(A + N_DIM * N_DIM + row * N_DIM, 0, 1);
      __builtin_prefetch(A + N_DIM * N_DIM + row * N_DIM + 128, 0, 1);
    }

#pragma unroll 4
    for (int kb = 0; kb < 64; ++kb) {
      const int k0 = kb * 4 + half * 2;
      const v2f a = *reinterpret_cast<const v2f*>(A + row * N_DIM + k0);
#pragma unroll
      for (int j = 0; j < 2; ++j) {
        const int b = (nt0 + j) * 16 + lm;
        const v2f bb = *reinterpret_cast<const v2f*>(&S[b * N_DIM + k0]);
        acc[j] = wmma_f32_16x16x4(a, bb, acc[j]);
      }
    }
    __syncthreads();
#pragma unroll
    for (int j = 0; j < 2; ++j) {
      const int b = (nt0 + j) * 16 + lm;
      store8(out + ((size_t)(t * B_DIM + b)) * N_DIM + nb, acc[j]);
      store8(&S[b * N_DIM + nb], acc[j]);
    }
    __syncthreads();
  }
}

extern "C" void kernel_launch(void* const* d_in, const int* in_sizes, int n_in,
                              void* d_out, int out_size, void* d_ws,
                              size_t ws_size, hipStream_t stream) {
  (void)in_sizes; (void)n_in; (void)out_size; (void)ws_size;
  const float* inp = (const float*)d_in[0];   // (L, B)
  const float* A_st = (const float*)d_in[1];  // (L, N, N)
  const float* B_st = (const float*)d_in[2];  // (L, N)
  float* out = (float*)d_out;                 // (L, B, N)

  // Workspace: P (64 x 256x256) | e (64 x 64x256) | s (64 x 64x256) = 24 MB
  float* P_ws = (float*)d_ws;
  float* e_ws = P_ws + (size_t)N_CHUNKS * N_DIM * N_DIM;
  float* s_ws = e_ws + (size_t)N_CHUNKS * B_DIM * N_DIM;

  hippo_pass1<<<N_CHUNKS * 2, 1024, 0, stream>>>(A_st, B_st, inp, P_ws, e_ws);
  hippo_pass2<<<1, 1024, 0, stream>>>(P_ws, e_ws, s_ws);
  hippo_pass3<<<N_CHUNKS, 1024, 0, stream>>>(A_st, B_st, inp, s_ws, out);
}